// BERTEmbedding_73237782331424
// MI455X (gfx1250) — compile-verified
//
#include <hip/hip_runtime.h>

typedef __attribute__((ext_vector_type(2))) float v2f;
typedef __attribute__((ext_vector_type(4))) float v4f;
typedef __attribute__((ext_vector_type(8))) float v8f;

#define NB   1024
#define NS   128
#define NF   10
#define ND   256
#define NROWS (NB * NS)          // 131072
#define OSTR  (2 * ND)           // 512 floats per output row

// -2*log2(10000)/256  : div_i = 10000^(-2i/256) = exp2(C0 * i)
#define PE_C0 (-0.103810253f)

// One block = 16 rows of the output. 512 threads = 16 waves (wave32).
// Wave w computes the 16x16 f32 WMMA tile C[m0..m0+15, 16w..16w+15] of
// obs_embed = X*W^T + b, then each wave also fills PE for one row.
__global__ __launch_bounds__(512) void fused_embed_kernel(
    const float* __restrict__ X,     // [NROWS, NF]
    const int*   __restrict__ doy,   // [NROWS]
    const float* __restrict__ W,     // [ND, NF]
    const float* __restrict__ bias,  // [ND]
    float*       __restrict__ out)   // [NROWS, OSTR]
{
    const int tid  = threadIdx.x;
    const int lane = tid & 31;
    const int wave = tid >> 5;          // 0..15 -> N tile
    const int m0   = blockIdx.x * 16;   // row tile base
    const int n0   = wave * 16;         // col tile base
    const int half = lane >> 4;         // 0: K pair {k,k+1}, 1: {k+2,k+3}
    const int l15  = lane & 15;         // M index (A) / N index (B)

    // ---------------- GEMM half: C = A(16x10) * B(10x16), K padded to 12 ----
    v8f c = {};
#pragma unroll
    for (int s = 0; s < 3; ++s) {
        const int k     = s * 4 + half * 2;        // lane-half's K pair base
        const int valid = (k < NF);                // pair {k,k+1}; k in {0,2,4,6,8,10}
        const int kc    = valid ? k : 0;
        const float msk = valid ? 1.0f : 0.0f;

        // A tile: 32-bit A 16x4 layout. lane l15 = row M, VGPR0/1 = K pair.
        v2f a = *(const v2f*)(X + (m0 + l15) * NF + kc);
        a *= msk;
        // B tile: mirrored layout, lane l15 = col N. B[k][n] = W[n*NF + k].
        v2f bm = *(const v2f*)(W + (n0 + l15) * NF + kc);
        bm *= msk;

        // v_wmma_f32_16x16x4_f32: (neg_a, A, neg_b, B, c_mod, C, reuse_a, reuse_b)
        c = __builtin_amdgcn_wmma_f32_16x16x4_f32(
                false, a, false, bm, (short)0, c, false, false);
    }

    // C/D layout: VGPR r -> M = r + 8*half, N = l15. Bias depends on N only.
    const float bv = bias[n0 + l15];
#pragma unroll
    for (int r = 0; r < 8; ++r) {
        const int row = m0 + r + half * 8;
        out[row * OSTR + n0 + l15] = c[r] + bv;
    }

    // ---------------- PE half: one row per wave, 8 floats per lane ----------
    // out[row, 256 + 2i]   = sin(p * div_i)
    // out[row, 256 + 2i+1] = cos(p * div_i),  div_i = exp2(PE_C0 * i)
    const int prow = m0 + wave;
    const float pf = (float)doy[prow];
    float* o = out + prow * OSTR + ND + lane * 8;

    v4f q0, q1;
#pragma unroll
    for (int j = 0; j < 4; ++j) {
        const float fi  = (float)(lane * 4 + j);
        const float ang = pf * __builtin_amdgcn_exp2f(PE_C0 * fi);  // v_exp_f32
        const float sv  = __sinf(ang);
        const float cv  = __cosf(ang);
        if (j < 2) { q0[2 * j] = sv; q0[2 * j + 1] = cv; }
        else       { q1[2 * (j - 2)] = sv; q1[2 * (j - 2) + 1] = cv; }
    }
    *(v4f*)(o)     = q0;   // global_store_b128
    *(v4f*)(o + 4) = q1;   // global_store_b128
}

extern "C" void kernel_launch(void* const* d_in, const int* in_sizes, int n_in,
                              void* d_out, int out_size, void* d_ws, size_t ws_size,
                              hipStream_t stream) {
    (void)in_sizes; (void)n_in; (void)out_size; (void)d_ws; (void)ws_size;
    const float* X    = (const float*)d_in[0];   // [1024,128,10] f32
    const int*   doy  = (const int*)  d_in[1];   // [1024,128]   i32
    const float* W    = (const float*)d_in[2];   // [256,10]     f32
    const float* bias = (const float*)d_in[3];   // [256]        f32
    float* out = (float*)d_out;                  // [1024,128,512] f32

    dim3 grid(NROWS / 16);   // 8192 blocks
    dim3 block(512);         // 16 wave32 waves
    fused_embed_kernel<<<grid, block, 0, stream>>>(X, doy, W, bias, out);
}